// GAT_9345848836281
// MI455X (gfx1250) — compile-verified
//
#include <hip/hip_runtime.h>

typedef __attribute__((ext_vector_type(2))) float v2f;
typedef __attribute__((ext_vector_type(8))) float v8f;

#define NNODES 20000

__device__ __forceinline__ float elu1(float v) { return v > 0.f ? v : __expf(v) - 1.f; }

// ---------------------------------------------------------------------------
// FP32 WMMA GEMM: C[M x NOUT] = act(A[M x 256]) @ B[256 x NOUT]
// One block = 16 output rows. 8 waves, each wave owns 16-wide N tiles.
// A tile staged in LDS (stride 260 to avoid bank conflicts for the WMMA
// per-lane column reads). Uses V_WMMA_F32_16X16X4_F32 (CDNA5).
// A-fragment layout (ISA 7.12.2): lanes 0-15 hold row M=lane, K=k..k+1 in
// v0/v1; lanes 16-31 hold K=k+2..k+3. B symmetric: lane = N column.
// C/D: VGPR i -> rows (i, i+8) at column lane&15.
// ---------------------------------------------------------------------------
template<int NOUT, bool ELU_A>
__global__ __launch_bounds__(256) void gemm_wmma_kernel(const float* __restrict__ A,
                                                        const float* __restrict__ B,
                                                        float* __restrict__ C) {
  constexpr int K   = 256;
  constexpr int LDA = 260;                 // 260 % 64 = 4 -> conflict-free lanes
  __shared__ float As[16 * LDA];
  const int m0  = blockIdx.x * 16;
  const int tid = threadIdx.x;

  // cooperative load of 16x256 A tile as float4 (global b128 -> ds b128)
#pragma unroll
  for (int i = 0; i < 4; ++i) {
    int idx = tid + i * 256;               // 0..1023 float4 slots
    int row = idx >> 6;                    // 64 float4 per row
    int c4  = (idx & 63) << 2;
    float4 v = *reinterpret_cast<const float4*>(A + (size_t)(m0 + row) * K + c4);
    if (ELU_A) { v.x = elu1(v.x); v.y = elu1(v.y); v.z = elu1(v.z); v.w = elu1(v.w); }
    *reinterpret_cast<float4*>(&As[row * LDA + c4]) = v;
  }
  __syncthreads();

  const int lane  = tid & 31;
  const int wave  = tid >> 5;
  const int arow  = lane & 15;             // A row / B,C column within tile
  const int khalf = (lane >> 4) << 1;      // 0 or 2

  for (int t = wave; t < NOUT / 16; t += 8) {
    const int n0 = t * 16;
    v8f acc = {};
#pragma unroll 8
    for (int k = 0; k < K; k += 4) {
      const int kk = k + khalf;
      v2f a, b;
      a.x = As[arow * LDA + kk];
      a.y = As[arow * LDA + kk + 1];
      b.x = B[(size_t)kk * NOUT + n0 + arow];
      b.y = B[(size_t)(kk + 1) * NOUT + n0 + arow];
      acc = __builtin_amdgcn_wmma_f32_16x16x4_f32(false, a, false, b,
                                                  (short)0, acc, false, false);
    }
    const int rbase = m0 + ((lane >> 4) << 3);   // +0 or +8
#pragma unroll
    for (int i = 0; i < 8; ++i)
      C[(size_t)(rbase + i) * NOUT + n0 + arow] = acc[i];
  }
}

// per-(node,head) attention dots: as[n,h]=<h[n,h,:],att_src[h,:]>, ad likewise
__global__ void alphas_kernel(const float* __restrict__ h,
                              const float* __restrict__ att_s,
                              const float* __restrict__ att_d,
                              float* __restrict__ out_s, float* __restrict__ out_d,
                              int N, int H, int Cc) {
  int t = blockIdx.x * blockDim.x + threadIdx.x;
  if (t >= N * H) return;
  int n = t / H, hh = t - n * H;
  const float* hp = h + (size_t)n * H * Cc + (size_t)hh * Cc;
  const float* sp = att_s + hh * Cc;
  const float* dp = att_d + hh * Cc;
  float s = 0.f, d = 0.f;
  for (int c = 0; c < Cc; ++c) { float v = hp[c]; s += v * sp[c]; d += v * dp[c]; }
  out_s[t] = s; out_d[t] = d;
}

__device__ __forceinline__ void edge_sd(const long long* __restrict__ ei, int eid,
                                        int E, int& s, int& d) {
  if (eid < E) { s = (int)ei[eid]; d = (int)ei[(size_t)E + eid]; }
  else         { s = d = eid - E; }                   // self-loop
}

// float atomic max via sign-split integer atomics
__device__ __forceinline__ void atomicMaxF(float* addr, float val) {
  if (val >= 0.f) atomicMax((int*)addr, __float_as_int(val));
  else            atomicMin((unsigned int*)addr, __float_as_uint(val));
}

__global__ void edge_logit_kernel(const long long* __restrict__ ei, int E, int Etot,
                                  const float* __restrict__ as_, const float* __restrict__ ad_,
                                  float* __restrict__ e, float* __restrict__ m, int H) {
  int t = blockIdx.x * blockDim.x + threadIdx.x;
  if (t >= Etot * H) return;
  int eid = t / H, hh = t - eid * H;
  int s, d; edge_sd(ei, eid, E, s, d);
  float v = as_[s * H + hh] + ad_[d * H + hh];
  v = v > 0.f ? v : 0.2f * v;                         // leaky_relu 0.2
  e[t] = v;
  atomicMaxF(&m[d * H + hh], v);
}

__global__ void edge_exp_kernel(const long long* __restrict__ ei, int E, int Etot,
                                float* __restrict__ e, const float* __restrict__ m,
                                float* __restrict__ den, int H) {
  int t = blockIdx.x * blockDim.x + threadIdx.x;
  if (t >= Etot * H) return;
  int eid = t / H, hh = t - eid * H;
  int s, d; edge_sd(ei, eid, E, s, d);
  float ex = __expf(e[t] - m[d * H + hh]);
  e[t] = ex;
  atomicAdd(&den[d * H + hh], ex);
}

// thread per (edge, h*C + c): coalesced row gather from h[src], coalesced
// atomicAdd burst into out[dst]
__global__ void edge_scatter_kernel(const long long* __restrict__ ei, int E, int Etot,
                                    const float* __restrict__ ex, const float* __restrict__ den,
                                    const float* __restrict__ h, float* __restrict__ out,
                                    int H, int Cc) {
  int HC = H * Cc;
  int t = blockIdx.x * blockDim.x + threadIdx.x;
  if (t >= Etot * HC) return;
  int eid = t / HC, r = t - eid * HC;
  int hh = r / Cc;
  int s, d; edge_sd(ei, eid, E, s, d);
  float a = ex[eid * H + hh] / (den[d * H + hh] + 1e-16f);
  atomicAdd(&out[(size_t)d * HC + r], h[(size_t)s * HC + r] * a);
}

__global__ void fill_kernel(float* __restrict__ p, float v, int n) {
  int t = blockIdx.x * blockDim.x + threadIdx.x;
  if (t < n) p[t] = v;
}

__global__ void bias_init_kernel(float* __restrict__ p, const float* __restrict__ b,
                                 int N, int Cc) {
  int t = blockIdx.x * blockDim.x + threadIdx.x;
  if (t >= N * Cc) return;
  p[t] = b[t % Cc];
}

static inline int cdiv(long long a, int b) { return (int)((a + b - 1) / b); }

extern "C" void kernel_launch(void* const* d_in, const int* in_sizes, int n_in,
                              void* d_out, int out_size, void* d_ws, size_t ws_size,
                              hipStream_t stream) {
  const float*     x   = (const float*)d_in[0];
  const long long* ei  = (const long long*)d_in[1];
  const float*     W1  = (const float*)d_in[2];
  const float*     aS1 = (const float*)d_in[3];
  const float*     aD1 = (const float*)d_in[4];
  const float*     b1  = (const float*)d_in[5];
  const float*     W2  = (const float*)d_in[6];
  const float*     aS2 = (const float*)d_in[7];
  const float*     aD2 = (const float*)d_in[8];
  const float*     b2  = (const float*)d_in[9];
  float* out = (float*)d_out;

  const int N = NNODES;
  const int E = in_sizes[1] / 2;
  const int Etot = E + N;

  float* ws   = (float*)d_ws;
  float* h1   = ws;                        // N*256
  float* agg1 = h1   + (size_t)N * 256;    // N*256
  float* e1   = agg1 + (size_t)N * 256;    // Etot*8
  float* m1   = e1   + (size_t)Etot * 8;   // N*8
  float* den1 = m1   + (size_t)N * 8;      // N*8
  float* als1 = den1 + (size_t)N * 8;      // N*8
  float* ald1 = als1 + (size_t)N * 8;      // N*8
  float* h2   = ald1 + (size_t)N * 8;      // N*64
  float* e2   = h2   + (size_t)N * 64;     // Etot
  float* m2   = e2   + (size_t)Etot;       // N
  float* den2 = m2   + (size_t)N;          // N
  float* als2 = den2 + (size_t)N;          // N
  float* ald2 = als2 + (size_t)N;          // N

  const int B = 256;
  const float NEG = -3.402823466e38f;

  // ---- layer 1 ----
  gemm_wmma_kernel<256, false><<<N / 16, 256, 0, stream>>>(x, W1, h1);
  alphas_kernel<<<cdiv(N * 8, B), B, 0, stream>>>(h1, aS1, aD1, als1, ald1, N, 8, 32);
  fill_kernel<<<cdiv(N * 8, B), B, 0, stream>>>(m1, NEG, N * 8);
  fill_kernel<<<cdiv(N * 8, B), B, 0, stream>>>(den1, 0.f, N * 8);
  bias_init_kernel<<<cdiv(N * 256, B), B, 0, stream>>>(agg1, b1, N, 256);
  edge_logit_kernel<<<cdiv((long long)Etot * 8, B), B, 0, stream>>>(ei, E, Etot, als1, ald1, e1, m1, 8);
  edge_exp_kernel<<<cdiv((long long)Etot * 8, B), B, 0, stream>>>(ei, E, Etot, e1, m1, den1, 8);
  edge_scatter_kernel<<<cdiv((long long)Etot * 256, B), B, 0, stream>>>(ei, E, Etot, e1, den1, h1, agg1, 8, 32);

  // ---- layer 2 (ELU fused into GEMM A-load) ----
  gemm_wmma_kernel<64, true><<<N / 16, 256, 0, stream>>>(agg1, W2, h2);
  alphas_kernel<<<cdiv(N, B), B, 0, stream>>>(h2, aS2, aD2, als2, ald2, N, 1, 64);
  fill_kernel<<<cdiv(N, B), B, 0, stream>>>(m2, NEG, N);
  fill_kernel<<<cdiv(N, B), B, 0, stream>>>(den2, 0.f, N);
  bias_init_kernel<<<cdiv(N * 64, B), B, 0, stream>>>(out, b2, N, 64);
  edge_logit_kernel<<<cdiv((long long)Etot, B), B, 0, stream>>>(ei, E, Etot, als2, ald2, e2, m2, 1);
  edge_exp_kernel<<<cdiv((long long)Etot, B), B, 0, stream>>>(ei, E, Etot, e2, m2, den2, 1);
  edge_scatter_kernel<<<cdiv((long long)Etot * 64, B), B, 0, stream>>>(ei, E, Etot, e2, den2, h2, out, 1, 64);
}